// WeightedMSELoss_28750511079907
// MI455X (gfx1250) — compile-verified
//
#include <hip/hip_runtime.h>
#include <math.h>

// Problem constants (match reference)
#define NROWS 4096
#define NCOLS 4096
#define TPB   256
#define K_TOP 5
#define TOPW  3.0f

typedef __attribute__((ext_vector_type(4))) float v4f;
typedef __attribute__((ext_vector_type(2))) float v2f;
typedef __attribute__((ext_vector_type(8))) float v8f;

// ---------------------------------------------------------------------------
// Kernel A: one block per row. Streams 2 rows (32 KiB) with NT b128 loads,
// computes rowSum(loss) and the sum of loss at the row's top-5 targets via
// 5 rounds of block-wide argmax (wave32 butterfly + LDS combine).
// ---------------------------------------------------------------------------
__global__ __launch_bounds__(TPB) void wmse_row_kernel(
    const float* __restrict__ preds,
    const float* __restrict__ targs,
    float* __restrict__ rowOut)
{
    __shared__ float s_sum[8];
    __shared__ float s_mv[8];
    __shared__ float s_ml[8];

    const int row  = blockIdx.x;
    const int tid  = threadIdx.x;
    const int lane = tid & 31;
    const int wave = tid >> 5;

    const v4f* p4 = reinterpret_cast<const v4f*>(preds) + (size_t)row * (NCOLS / 4);
    const v4f* t4 = reinterpret_cast<const v4f*>(targs) + (size_t)row * (NCOLS / 4);

    float tv[16], tl[16];
    float sum = 0.0f;

    #pragma unroll
    for (int c = 0; c < 4; ++c) {
        // single-use streaming data: non-temporal 128-bit loads
        v4f p = __builtin_nontemporal_load(&p4[c * TPB + tid]);
        v4f t = __builtin_nontemporal_load(&t4[c * TPB + tid]);
        float d0 = p.x - t.x, d1 = p.y - t.y, d2 = p.z - t.z, d3 = p.w - t.w;
        float l0 = d0 * d0, l1 = d1 * d1, l2 = d2 * d2, l3 = d3 * d3;
        sum += (l0 + l1) + (l2 + l3);
        tv[c * 4 + 0] = t.x; tl[c * 4 + 0] = l0;
        tv[c * 4 + 1] = t.y; tl[c * 4 + 1] = l1;
        tv[c * 4 + 2] = t.z; tl[c * 4 + 2] = l2;
        tv[c * 4 + 3] = t.w; tl[c * 4 + 3] = l3;
    }

    // ---- block sum of loss (wave32 butterfly, then 8-wave LDS combine) ----
    #pragma unroll
    for (int off = 16; off; off >>= 1)
        sum += __shfl_xor(sum, off, 32);
    if (lane == 0) s_sum[wave] = sum;
    __syncthreads();
    float rowSum = 0.0f;
    #pragma unroll
    for (int i = 0; i < 8; ++i) rowSum += s_sum[i];

    // ---- top-5 of targets: 5 rounds of block argmax, mask winner to -inf ----
    float topSum = 0.0f;
    for (int it = 0; it < K_TOP; ++it) {
        float mv = -INFINITY, ml = 0.0f;
        #pragma unroll
        for (int i = 0; i < 16; ++i) {
            if (tv[i] > mv) { mv = tv[i]; ml = tl[i]; }
        }
        // wave32 butterfly max carrying the paired loss value
        #pragma unroll
        for (int off = 16; off; off >>= 1) {
            float ov = __shfl_xor(mv, off, 32);
            float ol = __shfl_xor(ml, off, 32);
            if (ov > mv) { mv = ov; ml = ol; }
        }
        __syncthreads();                 // protect s_mv/s_ml reuse across rounds
        if (lane == 0) { s_mv[wave] = mv; s_ml[wave] = ml; }
        __syncthreads();
        float bmv = s_mv[0], bml = s_ml[0];
        #pragma unroll
        for (int i = 1; i < 8; ++i) {
            if (s_mv[i] > bmv) { bmv = s_mv[i]; bml = s_ml[i]; }
        }
        topSum += bml;
        // winner holds a bit-exact copy of bmv -> exact compare masks it out
        #pragma unroll
        for (int i = 0; i < 16; ++i) {
            if (tv[i] == bmv) tv[i] = -INFINITY;
        }
    }

    if (tid == 0) rowOut[row] = rowSum + (TOPW - 1.0f) * topSum;
}

// ---------------------------------------------------------------------------
// Kernel B: single wave32 reduces 4096 row partials to the scalar mean using
// V_WMMA_F32_16X16X4_F32 against an all-ones B matrix. D[m][n] += sum_k A[m][k],
// so the grand total over D counts every input exactly 16x (once per column n);
// divide by 16 at the end. f32 WMMA rounds RNE, same as scalar adds.
// ---------------------------------------------------------------------------
__global__ __launch_bounds__(32) void wmse_final_kernel(
    const float* __restrict__ rowSums,
    float* __restrict__ out)
{
    const int lane = threadIdx.x;        // 32 threads, EXEC all ones (required)

    v8f acc = {};
    v2f bones; bones.x = 1.0f; bones.y = 1.0f;

    for (int it = 0; it < NROWS / 64; ++it) {
        // Any bijection of 64 inputs onto the 16x4 A tile works for a total sum.
        v2f a;
        a.x = rowSums[it * 64 + lane];
        a.y = rowSums[it * 64 + 32 + lane];
        acc = __builtin_amdgcn_wmma_f32_16x16x4_f32(
            /*neg_a=*/false, a, /*neg_b=*/false, bones,
            /*c_mod=*/(short)0, acc, /*reuse_a=*/false, /*reuse_b=*/false);
    }

    float s = ((acc[0] + acc[1]) + (acc[2] + acc[3])) +
              ((acc[4] + acc[5]) + (acc[6] + acc[7]));
    #pragma unroll
    for (int off = 16; off; off >>= 1)
        s += __shfl_xor(s, off, 32);

    if (lane == 0) {
        const float invCnt = 1.0f / ((float)NROWS * (float)NCOLS);
        out[0] = s * (1.0f / 16.0f) * invCnt;
    }
}

// ---------------------------------------------------------------------------
extern "C" void kernel_launch(void* const* d_in, const int* in_sizes, int n_in,
                              void* d_out, int out_size, void* d_ws, size_t ws_size,
                              hipStream_t stream) {
    const float* preds = (const float*)d_in[0];
    const float* targs = (const float*)d_in[1];
    float* out    = (float*)d_out;
    float* rowOut = (float*)d_ws;    // 4096 floats of scratch, fully overwritten

    wmse_row_kernel<<<NROWS, TPB, 0, stream>>>(preds, targs, rowOut);
    wmse_final_kernel<<<1, 32, 0, stream>>>(rowOut, out);
}